// MotionLoss_62156766708038
// MI455X (gfx1250) — compile-verified
//
#include <hip/hip_runtime.h>
#include <hip/hip_bf16.h>

// ---------------------------------------------------------------------------
// MotionLoss for MI455X (gfx1250, wave32).
// Memory-bound: 264 MB streamed once -> ~11us floor at 23.3 TB/s.
// All five loss terms fold (lambda/count applied once per thread) into ONE
// scalar accumulator (d_out[0]) via V_WMMA_F32_16X16X4_F32 wave reduction +
// LDS block reduction + one global atomicAdd per block.
// Kernel 1 additionally spills the hip delta rotations (computed anyway for
// FK) densely into d_ws so the prefix-product scan kernel reads 72B dense
// records instead of 24B-of-504B sparse rows (cacheline utilization fix).
// ---------------------------------------------------------------------------

#define B_ 64
#define S_ 4096

constexpr float W_ROOT = (float)(5.0 / ((double)B_ * S_ * 6.0));     // l_root_delta
constexpr float W_ACC  = (float)(5.0 / ((double)B_ * S_ * 9.0));     // l_root_accum
constexpr float W_ROT  = (float)(1.0 / ((double)B_ * S_ * 120.0));   // l_rot
constexpr float W_POS  = (float)(5.0 / ((double)B_ * S_ * 66.0));    // l_pos
constexpr float W_VEL  = (float)(1.0 / ((double)B_ * (S_ - 1) * 66.0)); // l_vel

constexpr int kParent[22] = {-1,0,1,2,3,4,3,6,7,8,3,10,11,12,1,14,15,16,1,18,19,20};

typedef float v2f __attribute__((ext_vector_type(2)));
typedef float v8f __attribute__((ext_vector_type(8)));

// ---- wave32 sum via V_WMMA_F32_16X16X4_F32 --------------------------------
// A (16x4): VGPR0 = x (lanes 0-15 -> K=0, lanes 16-31 -> K=2), VGPR1 = 0.
// B (4x16) = all ones.  D[i][j] = x_i + x_{i+16}.  Sum lane's 8 C regs, then
// one shfl_xor(16) gives the full 32-lane sum in every lane.
__device__ __forceinline__ float wave_reduce(float x) {
    v2f a; a[0] = x;    a[1] = 0.0f;
    v2f b; b[0] = 1.0f; b[1] = 1.0f;
    v8f c = {0.f,0.f,0.f,0.f,0.f,0.f,0.f,0.f};
    c = __builtin_amdgcn_wmma_f32_16x16x4_f32(false, a, false, b, (short)0, c,
                                              false, false);
    float t = ((c[0] + c[1]) + (c[2] + c[3])) + ((c[4] + c[5]) + (c[6] + c[7]));
    t += __shfl_xor(t, 16, 32);
    return t;
}

// ---- 3x3 helpers (row-major) ----------------------------------------------
__device__ __forceinline__ void mat3_mul(const float* a, const float* b, float* c) {
#pragma unroll
    for (int r = 0; r < 3; ++r)
#pragma unroll
        for (int cc = 0; cc < 3; ++cc)
            c[r*3+cc] = fmaf(a[r*3+0], b[0*3+cc],
                        fmaf(a[r*3+1], b[1*3+cc], a[r*3+2] * b[2*3+cc]));
}
__device__ __forceinline__ void mat3_copy(float* d, const float* s) {
#pragma unroll
    for (int j = 0; j < 9; ++j) d[j] = s[j];
}

// cont6d -> 3x3 rotation (columns b1,b2,b3), matching the reference.
__device__ __forceinline__ void cont6d(const float* d, float* R) {
    float ax = d[0], ay = d[1], az = d[2];
    float bx = d[3], by = d[4], bz = d[5];
    float n1 = sqrtf(fmaf(ax, ax, fmaf(ay, ay, az * az)));
    float i1 = 1.0f / fmaxf(n1, 1e-12f);
    float b1x = ax * i1, b1y = ay * i1, b1z = az * i1;
    float dt = fmaf(b1x, bx, fmaf(b1y, by, b1z * bz));
    float c2x = fmaf(-b1x, dt, bx), c2y = fmaf(-b1y, dt, by), c2z = fmaf(-b1z, dt, bz);
    float n2 = sqrtf(fmaf(c2x, c2x, fmaf(c2y, c2y, c2z * c2z)));
    float i2 = 1.0f / fmaxf(n2, 1e-12f);
    float b2x = c2x * i2, b2y = c2y * i2, b2z = c2z * i2;
    float b3x = b1y * b2z - b1z * b2y;
    float b3y = b1z * b2x - b1x * b2z;
    float b3z = b1x * b2y - b1y * b2x;
    R[0] = b1x; R[1] = b2x; R[2] = b3x;
    R[3] = b1y; R[4] = b2y; R[5] = b3y;
    R[6] = b1z; R[7] = b2z; R[8] = b3z;
}

__device__ __forceinline__ void load6(const float* p, float* d) {
    // rows are 8B aligned (504B stride, 24B joint stride) -> b64 loads
    const float2* p2 = reinterpret_cast<const float2*>(p);
    float2 u0 = p2[0], u1 = p2[1], u2 = p2[2];
    d[0] = u0.x; d[1] = u0.y; d[2] = u1.x; d[3] = u1.y; d[4] = u2.x; d[5] = u2.y;
}

// load 18 consecutive floats (8B aligned) -> Rp[9], Rg[9]
__device__ __forceinline__ void load18(const float* p, float* Rp, float* Rg) {
    const float2* p2 = reinterpret_cast<const float2*>(p);
    float2 u[9];
#pragma unroll
    for (int j = 0; j < 9; ++j) u[j] = p2[j];
    Rp[0]=u[0].x; Rp[1]=u[0].y; Rp[2]=u[1].x; Rp[3]=u[1].y; Rp[4]=u[2].x;
    Rp[5]=u[2].y; Rp[6]=u[3].x; Rp[7]=u[3].y; Rp[8]=u[4].x;
    Rg[0]=u[4].y; Rg[1]=u[5].x; Rg[2]=u[5].y; Rg[3]=u[6].x; Rg[4]=u[6].y;
    Rg[5]=u[7].x; Rg[6]=u[7].y; Rg[7]=u[8].x; Rg[8]=u[8].y;
}

struct Sk { float R[9]; float p[3]; };

// FK over the 22-joint tree for one (pred,target) row pair.
// MODE 0: accumulate unweighted root(a0)/rot(a1)/pos(a2) sums, write
//         e = posP - posG to e_store, optionally spill hip rotations.
// MODE 1: accumulate unweighted vel sum into a0:  sum (e_cmp[j] - e[j])^2.
template <int MODE>
__device__ __forceinline__ void fk_pair(const float* __restrict__ P,
                                        const float* __restrict__ T,
                                        const float* __restrict__ offs,
                                        float* __restrict__ e_store,
                                        const float* __restrict__ e_cmp,
                                        float* __restrict__ rot_out,
                                        float& a0, float& a1, float& a2) {
    Sk cp, cg, a1p, a1g, a3p, a3g;
#pragma unroll
    for (int i = 1; i <= 21; ++i) {
        const int par = kParent[i];
        Sk Pp, Pg;
        if (i == 1) {
#pragma unroll
            for (int j = 0; j < 9; ++j) { Pp.R[j] = (j % 4 == 0) ? 1.f : 0.f; Pg.R[j] = Pp.R[j]; }
            Pp.p[0] = Pp.p[1] = Pp.p[2] = 0.f;
            Pg.p[0] = Pg.p[1] = Pg.p[2] = 0.f;
        } else if (par == 1) { Pp = a1p; Pg = a1g; }
        else if (par == 3)   { Pp = a3p; Pg = a3g; }
        else                 { Pp = cp;  Pg = cg;  }

        float dp[6], dg[6];
        load6(P + (i - 1) * 6, dp);
        load6(T + (i - 1) * 6, dg);
        if constexpr (MODE == 0) {
#pragma unroll
            for (int c2 = 0; c2 < 6; ++c2) {
                float d = dp[c2] - dg[c2];
                if (i == 1) a0 = fmaf(d, d, a0);   // root-delta term
                else        a1 = fmaf(d, d, a1);   // rot term
            }
        }
        float Rp[9], Rg[9];
        cont6d(dp, Rp);
        cont6d(dg, Rg);
        if constexpr (MODE == 0) {
            if (i == 1 && rot_out) {   // spill hip deltas for the scan kernel
                float2* o2 = reinterpret_cast<float2*>(rot_out);
                o2[0] = make_float2(Rp[0], Rp[1]);
                o2[1] = make_float2(Rp[2], Rp[3]);
                o2[2] = make_float2(Rp[4], Rp[5]);
                o2[3] = make_float2(Rp[6], Rp[7]);
                o2[4] = make_float2(Rp[8], Rg[0]);
                o2[5] = make_float2(Rg[1], Rg[2]);
                o2[6] = make_float2(Rg[3], Rg[4]);
                o2[7] = make_float2(Rg[5], Rg[6]);
                o2[8] = make_float2(Rg[7], Rg[8]);
            }
        }
        Sk np, ng;
        mat3_mul(Pp.R, Rp, np.R);
        mat3_mul(Pg.R, Rg, ng.R);
        const float ox = offs[i*3+0], oy = offs[i*3+1], oz = offs[i*3+2];
#pragma unroll
        for (int r = 0; r < 3; ++r) {
            np.p[r] = Pp.p[r] + fmaf(Pp.R[r*3+0], ox, fmaf(Pp.R[r*3+1], oy, Pp.R[r*3+2] * oz));
            ng.p[r] = Pg.p[r] + fmaf(Pg.R[r*3+0], ox, fmaf(Pg.R[r*3+1], oy, Pg.R[r*3+2] * oz));
        }
#pragma unroll
        for (int r = 0; r < 3; ++r) {
            float e = np.p[r] - ng.p[r];
            if constexpr (MODE == 0) {
                a2 = fmaf(e, e, a2);               // pos term
                e_store[(i - 1) * 3 + r] = e;
            } else {
                float d = e_cmp[(i - 1) * 3 + r] - e;
                a0 = fmaf(d, d, a0);               // vel term
            }
        }
        cp = np; cg = ng;
        if (i == 1) { a1p = np; a1g = ng; }
        if (i == 3) { a3p = np; a3g = ng; }
    }
}

// ---------------------------------------------------------------------------
// Kernel 1: per-(b,s) thread: root-delta + rot + pos; e shared in LDS for vel.
// ---------------------------------------------------------------------------
template <bool SAVE_ROT>
__global__ __launch_bounds__(256) void fk_loss_kernel(const float* __restrict__ pred,
                                                      const float* __restrict__ tgt,
                                                      const float* __restrict__ offs,
                                                      float* __restrict__ out,
                                                      float* __restrict__ rot_ws) {
    __shared__ float eb[256 * 63];   // e per thread, stride 63 (bank-conflict-free)
    __shared__ float wsum[8];
    const int lt = threadIdx.x;
    const size_t gid = (size_t)blockIdx.x * 256 + lt;
    const int s = (int)(gid & (S_ - 1));
    const float* P = pred + gid * 126;
    const float* T = tgt + gid * 126;
    float* ro = SAVE_ROT ? (rot_ws + gid * 18) : nullptr;

    float a0 = 0.f, a1 = 0.f, a2 = 0.f, a3 = 0.f;
    fk_pair<0>(P, T, offs, &eb[lt * 63], nullptr, ro, a0, a1, a2);
    __syncthreads();

    if (s > 0) {
        if (lt > 0) {
#pragma unroll
            for (int j = 0; j < 63; ++j) {
                float d = eb[lt * 63 + j] - eb[(lt - 1) * 63 + j];
                a3 = fmaf(d, d, a3);
            }
        } else {
            // block-boundary lane recomputes e(s-1) on the fly
            float du0 = 0.f, du1 = 0.f;
            fk_pair<1>(P - 126, T - 126, offs, nullptr, &eb[0], nullptr, a3, du0, du1);
        }
    }

    // fold lambda/count once per thread, then reduce
    float acc = fmaf(W_ROOT, a0, fmaf(W_ROT, a1, fmaf(W_POS, a2, W_VEL * a3)));
    float tot = wave_reduce(acc);           // EXEC full here (uniform path)
    if ((lt & 31) == 0) wsum[lt >> 5] = tot;
    __syncthreads();
    if (lt == 0) {
        float v = 0.f;
#pragma unroll
        for (int w = 0; w < 8; ++w) v += wsum[w];
        atomicAdd(out, v);
    }
}

// ---------------------------------------------------------------------------
// Kernel 2 (compact): hip prefix products from dense 18-float records in ws.
// Block = batch row, 16 steps/thread, Hillis-Steele 3x3 scan in LDS.
// ---------------------------------------------------------------------------
__global__ __launch_bounds__(256) void hip_scan_compact_kernel(const float* __restrict__ rots,
                                                               float* __restrict__ out) {
    __shared__ float sp[256 * 9];
    __shared__ float sg[256 * 9];
    __shared__ float red[8];
    const int t = threadIdx.x;
    const float* Rb = rots + ((size_t)blockIdx.x * S_ + (size_t)t * 16) * 18;

    float Ap[9] = {1,0,0, 0,1,0, 0,0,1};
    float Ag[9] = {1,0,0, 0,1,0, 0,0,1};
    for (int k = 0; k < 16; ++k) {
        float Rp[9], Rg[9], tmp[9];
        load18(Rb + k * 18, Rp, Rg);
        mat3_mul(Ap, Rp, tmp); mat3_copy(Ap, tmp);
        mat3_mul(Ag, Rg, tmp); mat3_copy(Ag, tmp);
    }
#pragma unroll
    for (int j = 0; j < 9; ++j) { sp[t * 9 + j] = Ap[j]; sg[t * 9 + j] = Ag[j]; }
    __syncthreads();

    for (int d = 1; d < 256; d <<= 1) {
        float lp[9], lg[9], mp[9], mg[9];
#pragma unroll
        for (int j = 0; j < 9; ++j) { mp[j] = sp[t * 9 + j]; mg[j] = sg[t * 9 + j]; }
        const bool has = (t >= d);
        if (has) {
#pragma unroll
            for (int j = 0; j < 9; ++j) { lp[j] = sp[(t - d) * 9 + j]; lg[j] = sg[(t - d) * 9 + j]; }
        }
        __syncthreads();
        if (has) {
            float r9[9];
            mat3_mul(lp, mp, r9);
#pragma unroll
            for (int j = 0; j < 9; ++j) sp[t * 9 + j] = r9[j];
            mat3_mul(lg, mg, r9);
#pragma unroll
            for (int j = 0; j < 9; ++j) sg[t * 9 + j] = r9[j];
        }
        __syncthreads();
    }

    float Ep[9] = {1,0,0, 0,1,0, 0,0,1};
    float Eg[9] = {1,0,0, 0,1,0, 0,0,1};
    if (t > 0) {
#pragma unroll
        for (int j = 0; j < 9; ++j) { Ep[j] = sp[(t - 1) * 9 + j]; Eg[j] = sg[(t - 1) * 9 + j]; }
    }
    float acc = 0.f;
    for (int k = 0; k < 16; ++k) {
        float Rp[9], Rg[9], tmp[9];
        load18(Rb + k * 18, Rp, Rg);
        mat3_mul(Ep, Rp, tmp); mat3_copy(Ep, tmp);
        mat3_mul(Eg, Rg, tmp); mat3_copy(Eg, tmp);
#pragma unroll
        for (int j = 0; j < 9; ++j) {
            float dd = Ep[j] - Eg[j];
            acc = fmaf(dd, dd, acc);
        }
    }

    float tot = wave_reduce(W_ACC * acc);
    if ((t & 31) == 0) red[t >> 5] = tot;
    __syncthreads();
    if (t == 0) {
        float v = 0.f;
#pragma unroll
        for (int w = 0; w < 8; ++w) v += red[w];
        atomicAdd(out, v);
    }
}

// ---------------------------------------------------------------------------
// Kernel 2 (fallback, ws too small): same scan reading raw 6d rows.
// ---------------------------------------------------------------------------
__global__ __launch_bounds__(256) void hip_scan_kernel(const float* __restrict__ pred,
                                                       const float* __restrict__ tgt,
                                                       float* __restrict__ out) {
    __shared__ float sp[256 * 9];
    __shared__ float sg[256 * 9];
    __shared__ float red[8];
    const int t = threadIdx.x;
    const size_t base = ((size_t)blockIdx.x * S_ + (size_t)t * 16) * 126;
    const float* Pb = pred + base;
    const float* Tb = tgt + base;

    float Ap[9] = {1,0,0, 0,1,0, 0,0,1};
    float Ag[9] = {1,0,0, 0,1,0, 0,0,1};
    for (int k = 0; k < 16; ++k) {
        float d6[6], R[9], tmp[9];
        load6(Pb + k * 126, d6); cont6d(d6, R);
        mat3_mul(Ap, R, tmp); mat3_copy(Ap, tmp);
        load6(Tb + k * 126, d6); cont6d(d6, R);
        mat3_mul(Ag, R, tmp); mat3_copy(Ag, tmp);
    }
#pragma unroll
    for (int j = 0; j < 9; ++j) { sp[t * 9 + j] = Ap[j]; sg[t * 9 + j] = Ag[j]; }
    __syncthreads();

    for (int d = 1; d < 256; d <<= 1) {
        float lp[9], lg[9], mp[9], mg[9];
#pragma unroll
        for (int j = 0; j < 9; ++j) { mp[j] = sp[t * 9 + j]; mg[j] = sg[t * 9 + j]; }
        const bool has = (t >= d);
        if (has) {
#pragma unroll
            for (int j = 0; j < 9; ++j) { lp[j] = sp[(t - d) * 9 + j]; lg[j] = sg[(t - d) * 9 + j]; }
        }
        __syncthreads();
        if (has) {
            float r9[9];
            mat3_mul(lp, mp, r9);
#pragma unroll
            for (int j = 0; j < 9; ++j) sp[t * 9 + j] = r9[j];
            mat3_mul(lg, mg, r9);
#pragma unroll
            for (int j = 0; j < 9; ++j) sg[t * 9 + j] = r9[j];
        }
        __syncthreads();
    }

    float Ep[9] = {1,0,0, 0,1,0, 0,0,1};
    float Eg[9] = {1,0,0, 0,1,0, 0,0,1};
    if (t > 0) {
#pragma unroll
        for (int j = 0; j < 9; ++j) { Ep[j] = sp[(t - 1) * 9 + j]; Eg[j] = sg[(t - 1) * 9 + j]; }
    }
    float acc = 0.f;
    for (int k = 0; k < 16; ++k) {
        float d6[6], R[9], tmp[9];
        load6(Pb + k * 126, d6); cont6d(d6, R);
        mat3_mul(Ep, R, tmp); mat3_copy(Ep, tmp);
        load6(Tb + k * 126, d6); cont6d(d6, R);
        mat3_mul(Eg, R, tmp); mat3_copy(Eg, tmp);
#pragma unroll
        for (int j = 0; j < 9; ++j) {
            float dd = Ep[j] - Eg[j];
            acc = fmaf(dd, dd, acc);
        }
    }

    float tot = wave_reduce(W_ACC * acc);
    if ((t & 31) == 0) red[t >> 5] = tot;
    __syncthreads();
    if (t == 0) {
        float v = 0.f;
#pragma unroll
        for (int w = 0; w < 8; ++w) v += red[w];
        atomicAdd(out, v);
    }
}

__global__ void zero_kernel(float* out) { out[0] = 0.0f; }

// ---------------------------------------------------------------------------
extern "C" void kernel_launch(void* const* d_in, const int* in_sizes, int n_in,
                              void* d_out, int out_size, void* d_ws, size_t ws_size,
                              hipStream_t stream) {
    (void)in_sizes; (void)n_in; (void)out_size;
    const float* pred = (const float*)d_in[0];
    const float* tgt  = (const float*)d_in[1];
    const float* offs = (const float*)d_in[2];
    float* out = (float*)d_out;

    const size_t need = (size_t)B_ * S_ * 18 * sizeof(float);  // 18.9 MB
    const bool compact = (d_ws != nullptr) && (ws_size >= need);

    zero_kernel<<<1, 1, 0, stream>>>(out);
    if (compact) {
        fk_loss_kernel<true><<<(B_ * S_) / 256, 256, 0, stream>>>(pred, tgt, offs, out,
                                                                  (float*)d_ws);
        hip_scan_compact_kernel<<<B_, 256, 0, stream>>>((const float*)d_ws, out);
    } else {
        fk_loss_kernel<false><<<(B_ * S_) / 256, 256, 0, stream>>>(pred, tgt, offs, out,
                                                                   nullptr);
        hip_scan_kernel<<<B_, 256, 0, stream>>>(pred, tgt, out);
    }
}